// Local2DAttention_49787260895773
// MI455X (gfx1250) — compile-verified
//
#include <hip/hip_runtime.h>

typedef float v2f __attribute__((ext_vector_type(2)));
typedef float v8f __attribute__((ext_vector_type(8)));

#define HH 64
#define WW 64
#define DD 256
#define NP 112          // 10x10 patch (100 keys) padded to 7 N-tiles of 16
#define SSTRIDE 114     // even stride: 8B-aligned ds_load_b64, staggered banks

__global__ __launch_bounds__(128)
void local_attn_wmma_f32(const float* __restrict__ x, float* __restrict__ out) {
    __shared__ float sP[16 * SSTRIDE];   // scores -> probs, 16 x 112 (padded)
    __shared__ int   sKB[NP];            // per-patch-key global base offsets

    const int blk = blockIdx.x;
    const int b   = blk >> 8;            // 256 tiles per batch image
    const int t   = blk & 255;
    const int h0  = (t >> 4) << 2;       // 4x4 query tile origin
    const int w0  = (t & 15) << 2;

    const int tid  = threadIdx.x;
    const int wave = tid >> 5;
    const int lane = tid & 31;
    const int half = lane >> 4;          // K-pair selector for A/B operands
    const int lm   = lane & 15;

    // ---- Phase 0: patch key base-address table (clamped; OOB handled by mask)
    if (tid < NP) {
        int p  = tid < 100 ? tid : 99;
        int pi = p / 10, pj = p - pi * 10;
        int kh = h0 - 3 + pi; kh = kh < 0 ? 0 : (kh > HH - 1 ? HH - 1 : kh);
        int kw = w0 - 3 + pj; kw = kw < 0 ? 0 : (kw > WW - 1 ? WW - 1 : kw);
        sKB[tid] = ((b * HH + kh) * WW + kw) * DD;
    }
    __syncthreads();

    // ---- Phase 1: S[16 x 112] = Q[16 x 256] * K^T  (f32 WMMA, K=4 steps)
    {
        const int m  = lm;                                   // A row = M
        const int qh = h0 + (m >> 2), qw = w0 + (m & 3);
        const long qbase = (long)((b * HH + qh) * WW + qw) * DD + 2 * half;
        for (int tn = wave; tn < 7; tn += 4) {
            const int  n     = tn * 16 + lm;                 // B col = N
            const long kbase = (long)sKB[n] + 2 * half;
            v8f acc = {};
            #pragma unroll 8
            for (int d0 = 0; d0 < DD; d0 += 4) {
                v2f a  = *(const v2f*)(x + qbase + d0);
                v2f bb = *(const v2f*)(x + kbase + d0);
                acc = __builtin_amdgcn_wmma_f32_16x16x4_f32(
                        false, a, false, bb, (short)0, acc, false, false);
            }
            #pragma unroll
            for (int r = 0; r < 8; ++r)
                sP[(r + 8 * half) * SSTRIDE + tn * 16 + lm] = acc[r];
        }
    }
    __syncthreads();

    // ---- Phase 2: masked softmax per query row (16 lanes, rest idle)
    if (tid < 16) {
        const int m  = tid;
        const int qh = h0 + (m >> 2), qw = w0 + (m & 3);
        float* row = sP + m * SSTRIDE;
        float mx = -3.0e38f;
        for (int n = 0; n < NP; ++n) {
            int  pi = n / 10, pj = n - pi * 10;
            int  kh = h0 - 3 + pi, kw = w0 - 3 + pj;
            bool valid = (n < 100) && kh >= 0 && kh < HH && kw >= 0 && kw < WW
                         && (kh - qh) >= -3 && (kh - qh) <= 3
                         && (kw - qw) >= -3 && (kw - qw) <= 3;
            float s = row[n] * 0.0625f;          // * D^-0.5
            row[n]  = valid ? s : -3.0e38f;
            if (valid && s > mx) mx = s;
        }
        float sum = 0.f;
        for (int n = 0; n < NP; ++n) {
            float s = row[n];
            float e = (s > -1.0e38f) ? __expf(s - mx) : 0.f;
            row[n]  = e;
            sum += e;
        }
        float inv = 1.f / sum;
        for (int n = 0; n < NP; ++n) row[n] *= inv;
    }
    __syncthreads();

    // ---- Phase 3: O[16 x 256] = P[16 x 112] * V[112 x 256]
    {
        for (int dt = wave; dt < 16; dt += 4) {
            const int d = dt * 16 + lm;                      // B col = N = channel
            v8f acc = {};
            #pragma unroll 4
            for (int p0 = 0; p0 < NP; p0 += 4) {
                const int pk = p0 + 2 * half;                // K rows for this lane half
                v2f a = *(const v2f*)(sP + lm * SSTRIDE + pk);   // P[lm][pk..pk+1]
                v2f bb;
                bb.x = x[(long)sKB[pk]     + d];             // V[pk][d]
                bb.y = x[(long)sKB[pk + 1] + d];             // V[pk+1][d]
                acc = __builtin_amdgcn_wmma_f32_16x16x4_f32(
                        false, a, false, bb, (short)0, acc, false, false);
            }
            #pragma unroll
            for (int r = 0; r < 8; ++r) {
                const int rowm = r + 8 * half;
                const int qh = h0 + (rowm >> 2), qw = w0 + (rowm & 3);
                out[(long)((b * HH + qh) * WW + qw) * DD + d] = acc[r];
            }
        }
    }
}

extern "C" void kernel_launch(void* const* d_in, const int* in_sizes, int n_in,
                              void* d_out, int out_size, void* d_ws, size_t ws_size,
                              hipStream_t stream) {
    const float* x   = (const float*)d_in[0];
    float*       out = (float*)d_out;
    // B=4, H=W=64 -> 4 * 16 * 16 = 1024 tiles of 4x4 queries
    local_attn_wmma_f32<<<dim3(1024), dim3(128), 0, stream>>>(x, out);
}